// STAGNNAggregator_28080496181651
// MI455X (gfx1250) — compile-verified
//
#include <hip/hip_runtime.h>
#include <hip/hip_bf16.h>

typedef __attribute__((ext_vector_type(2))) float v2f;
typedef __attribute__((ext_vector_type(8))) float v8f;

#define TT 6
#define NN 10000
#define EE 50000
#define DD 64
#define HEADS 4
#define DH 16
#define WSTRIDE_K 72

// ---------------------------------------------------------------------------
// Ordered float <-> uint encoding so atomicMax(uint) implements float max
// (handles negative values correctly).
__device__ __forceinline__ unsigned encF(float f) {
    unsigned u = __float_as_uint(f);
    return (u & 0x80000000u) ? ~u : (u | 0x80000000u);
}
__device__ __forceinline__ float decF(unsigned k) {
    unsigned u = (k & 0x80000000u) ? (k & 0x7FFFFFFFu) : ~k;
    return __uint_as_float(u);
}

// ---------------------------------------------------------------------------
// Precompute K_pe[dt][64] = pe[dt] @ W_k[:,64:72]^T  and rel_bias[dt] = -log1p(dt)
__global__ void pe_kernel(const float* __restrict__ Wk,
                          float* __restrict__ Kpe,
                          float* __restrict__ RelBias) {
    const int n = threadIdx.x;  // 0..63
    for (int dt = 0; dt <= 8; ++dt) {
        const float fdt = (float)dt;
        float pe[8];
        pe[0] = expf(-fdt * 0.25f);    // tau=4
        pe[1] = expf(-fdt * 0.0625f);  // tau=16
        pe[2] = sinf(fdt * 1.0f);
        pe[3] = sinf(fdt * 0.5f);
        pe[4] = sinf(fdt * 0.25f);
        pe[5] = cosf(fdt * 1.0f);
        pe[6] = cosf(fdt * 0.5f);
        pe[7] = cosf(fdt * 0.25f);
        float acc = 0.0f;
#pragma unroll
        for (int j = 0; j < 8; ++j) acc += pe[j] * Wk[n * WSTRIDE_K + 64 + j];
        Kpe[dt * 64 + n] = acc;
        if (n == 0) RelBias[dt] = -log1pf(fdt);
    }
}

// ---------------------------------------------------------------------------
// Fp32 WMMA projection: Out[r, n] = sum_k H[r,k] * W[n,k]   (Out = H @ W^T)
// One wave -> one 16x16 output tile; K-loop of 16 x v_wmma_f32_16x16x4_f32.
// blockIdx.x = row tile (16 rows), blockIdx.y = 0:Q 1:K 2:V, wave = col tile.
__global__ __launch_bounds__(128) void proj_kernel(
    const float* __restrict__ H,
    const float* __restrict__ Wq, const float* __restrict__ Wk,
    const float* __restrict__ Wv,
    float* __restrict__ Qt, float* __restrict__ Kt, float* __restrict__ Vt) {
    const int lane = threadIdx.x & 31;
    const int wave = threadIdx.x >> 5;
    const int row0 = blockIdx.x * 16;
    const int which = blockIdx.y;

    const float* W; int wstride; float* Out;
    if (which == 0)      { W = Wq; wstride = 64;        Out = Qt; }
    else if (which == 1) { W = Wk; wstride = WSTRIDE_K; Out = Kt; }
    else                 { W = Wv; wstride = 64;        Out = Vt; }

    const int m  = lane & 15;          // row within tile (A), col within tile (B)
    const int kb = (lane >> 4) << 1;   // lanes 16-31 handle K+2,K+3
    const int n0 = wave * 16;

    const float* __restrict__ arow = H + (size_t)(row0 + m) * DD + kb;
    const float* __restrict__ brow = W + (size_t)(n0 + m) * wstride + kb;

    v8f c = {};
#pragma unroll
    for (int kk = 0; kk < DD; kk += 4) {
        v2f a, b;
        a.x = arow[kk]; a.y = arow[kk + 1];
        b.x = brow[kk]; b.y = brow[kk + 1];
        c = __builtin_amdgcn_wmma_f32_16x16x4_f32(false, a, false, b,
                                                  (short)0, c, false, false);
    }
    // D layout: VGPR v -> row M=v (lanes 0-15) / M=v+8 (lanes 16-31), col = lane&15
    const int mrow = row0 + (lane >> 4) * 8;
#pragma unroll
    for (int v = 0; v < 8; ++v)
        Out[(size_t)(mrow + v) * DD + n0 + m] = c[v];
}

// ---------------------------------------------------------------------------
// Score helper shared by both edge passes.
__device__ __forceinline__ float edge_score(const float* __restrict__ q,
                                            const float* __restrict__ Kt,
                                            const float* __restrict__ Kpe,
                                            const float* __restrict__ RelBias,
                                            const float* __restrict__ S,
                                            int tp, int dt, int s, int h) {
    const float* __restrict__ kp  = Kt + ((size_t)(tp * NN + s) * DD + h * DH);
    const float* __restrict__ kpe = Kpe + dt * 64 + h * DH;
    float dot = 0.0f;
#pragma unroll
    for (int c = 0; c < DH; ++c) dot += q[c] * (kp[c] + kpe[c]);
    const float gate = logf(S[tp * NN + s] + 1e-6f);
    return dot * 0.25f + RelBias[dt] + gate;  // scale = dh^-0.5 = 0.25, TEMP = 1
}

// ---------------------------------------------------------------------------
// Pass A: per-(t, edge, head) max score -> atomicMax into segment max table.
__global__ __launch_bounds__(256) void score_max_kernel(
    const float* __restrict__ Qt, const float* __restrict__ Kt,
    const float* __restrict__ Kpe, const float* __restrict__ RelBias,
    const float* __restrict__ S, const int* __restrict__ src,
    const int* __restrict__ dst, unsigned* __restrict__ MaxBuf) {
    const int tid = blockIdx.x * blockDim.x + threadIdx.x;
    if (tid >= TT * EE * HEADS) return;
    const int h = tid & 3;
    const int e = (tid >> 2) % EE;
    const int t = (tid >> 2) / EE;
    const int s = src[e];
    const int d = dst[e];

    float q[DH];
    const float* __restrict__ qp = Qt + ((size_t)(t * NN + d) * DD + h * DH);
#pragma unroll
    for (int c = 0; c < DH; ++c) q[c] = qp[c];

    float m = -3.0e38f;
    for (int dt = 0; dt <= t; ++dt) {  // t <= 5 < W=8: all dt<=t valid
        const float sc = edge_score(q, Kt, Kpe, RelBias, S, t - dt, dt, s, h);
        m = fmaxf(m, sc);
    }
    atomicMax(&MaxBuf[(size_t)(t * NN + d) * HEADS + h], encF(m));
}

// ---------------------------------------------------------------------------
// Pass B: exp(score - segmax), accumulate ex*V and denom via atomics.
__global__ __launch_bounds__(256) void accum_kernel(
    const float* __restrict__ Qt, const float* __restrict__ Kt,
    const float* __restrict__ Vt, const float* __restrict__ Kpe,
    const float* __restrict__ RelBias, const float* __restrict__ S,
    const int* __restrict__ src, const int* __restrict__ dst,
    const unsigned* __restrict__ MaxBuf, float* __restrict__ OutNumer,
    float* __restrict__ Denom) {
    const int tid = blockIdx.x * blockDim.x + threadIdx.x;
    if (tid >= TT * EE * HEADS) return;
    const int h = tid & 3;
    const int e = (tid >> 2) % EE;
    const int t = (tid >> 2) / EE;
    const int s = src[e];
    const int d = dst[e];

    float q[DH];
    const float* __restrict__ qp = Qt + ((size_t)(t * NN + d) * DD + h * DH);
#pragma unroll
    for (int c = 0; c < DH; ++c) q[c] = qp[c];

    const float m = decF(MaxBuf[(size_t)(t * NN + d) * HEADS + h]);

    float numer[DH];
#pragma unroll
    for (int c = 0; c < DH; ++c) numer[c] = 0.0f;
    float den = 0.0f;

    for (int dt = 0; dt <= t; ++dt) {
        const int tp = t - dt;
        const float sc = edge_score(q, Kt, Kpe, RelBias, S, tp, dt, s, h);
        const float ex = expf(sc - m);
        den += ex;
        const float* __restrict__ vp = Vt + ((size_t)(tp * NN + s) * DD + h * DH);
#pragma unroll
        for (int c = 0; c < DH; ++c) numer[c] += ex * vp[c];
    }

    float* __restrict__ op = OutNumer + ((size_t)(t * NN + d) * DD + h * DH);
#pragma unroll
    for (int c = 0; c < DH; ++c) atomicAdd(&op[c], numer[c]);
    atomicAdd(&Denom[(size_t)(t * NN + d) * HEADS + h], den);
}

// ---------------------------------------------------------------------------
// Finalize: out = numer / max(denom, 1e-12); empty segments stay 0.
__global__ __launch_bounds__(256) void finalize_kernel(
    float* __restrict__ out, const float* __restrict__ Denom) {
    const int tid = blockIdx.x * blockDim.x + threadIdx.x;
    if (tid >= TT * NN * DD) return;
    const int node = tid >> 6;        // (t*N + n)
    const int h = (tid >> 4) & 3;
    const float den = Denom[(size_t)node * HEADS + h];
    out[tid] = out[tid] / fmaxf(den, 1e-12f);
}

// ---------------------------------------------------------------------------
extern "C" void kernel_launch(void* const* d_in, const int* in_sizes, int n_in,
                              void* d_out, int out_size, void* d_ws, size_t ws_size,
                              hipStream_t stream) {
    const float* H  = (const float*)d_in[0];  // (T, N, 64)
    const float* S  = (const float*)d_in[1];  // (T, N)
    const float* Wq = (const float*)d_in[2];  // (64, 64)
    const float* Wk = (const float*)d_in[3];  // (64, 72)
    const float* Wv = (const float*)d_in[4];  // (64, 64)
    const int*   ei = (const int*)d_in[5];    // (2, E)
    (void)in_sizes; (void)n_in; (void)ws_size;

    const size_t TN = (size_t)TT * NN;

    // Workspace layout (all float-aligned):
    float* Qt      = (float*)d_ws;          // TN*64
    float* Kt      = Qt + TN * DD;          // TN*64
    float* Vt      = Kt + TN * DD;          // TN*64
    float* Kpe     = Vt + TN * DD;          // 9*64
    float* RelBias = Kpe + 9 * 64;          // 16 (9 used)
    unsigned* MaxBuf = (unsigned*)(RelBias + 16);  // TN*4
    float* Denom   = (float*)(MaxBuf + TN * HEADS); // TN*4
    float* out     = (float*)d_out;

    hipMemsetAsync(d_out, 0, (size_t)out_size * sizeof(float), stream);
    hipMemsetAsync(MaxBuf, 0, TN * HEADS * sizeof(unsigned), stream);
    hipMemsetAsync(Denom, 0, TN * HEADS * sizeof(float), stream);

    pe_kernel<<<1, 64, 0, stream>>>(Wk, Kpe, RelBias);

    dim3 pgrid((unsigned)(TN / 16), 3);  // 3750 row tiles x {Q,K,V}
    proj_kernel<<<pgrid, 128, 0, stream>>>(H, Wq, Wk, Wv, Qt, Kt, Vt);

    const int* src = ei;
    const int* dst = ei + EE;
    const int nth = TT * EE * HEADS;
    score_max_kernel<<<(nth + 255) / 256, 256, 0, stream>>>(
        Qt, Kt, Kpe, RelBias, S, src, dst, MaxBuf);
    accum_kernel<<<(nth + 255) / 256, 256, 0, stream>>>(
        Qt, Kt, Vt, Kpe, RelBias, S, src, dst, MaxBuf, out, Denom);

    const int total = TT * NN * DD;
    finalize_kernel<<<(total + 255) / 256, 256, 0, stream>>>(out, Denom);
}